// DistributionLoss_10651518894647
// MI455X (gfx1250) — compile-verified
//
#include <hip/hip_runtime.h>

// Problem constants (from reference setup_inputs)
#define N_IMG 16
#define C_CLS 19
#define H_DIM 512
#define W_DIM 1024
#define HW_PIX (H_DIM * W_DIM)          // 524288 pixels per image
#define THREADS 256
#define PIX_PER_THREAD 4
#define TILE (THREADS * PIX_PER_THREAD) // 1024 pixels per block
#define BLOCKS_PER_IMG (HW_PIX / TILE)  // 512
#define IGNORE_IDX 255u

// d_ws layout (uint32):
//   [0, N*C)        : pred histogram  (per image)
//   [N*C, 2*N*C)    : raw target histogram (per image, bins = raw target value)
//   [2*N*C]         : global min of target (for the reference's `shift`)
#define WS_PRED 0
#define WS_TGT  (N_IMG * C_CLS)
#define WS_MIN  (2 * N_IMG * C_CLS)

typedef float f32x4 __attribute__((ext_vector_type(4)));
typedef int   v4i   __attribute__((vector_size(16)));   // matches builtin param pointee
#define AS_GLOBAL __attribute__((address_space(1)))
#define AS_LDS    __attribute__((address_space(3)))

#if defined(__gfx1250__) && __has_builtin(__builtin_amdgcn_global_load_async_to_lds_b128)
#define HAVE_ASYNC_LDS 1
#else
#define HAVE_ASYNC_LDS 0
#endif

__global__ void dl_init_ws(unsigned int* __restrict__ ws) {
    int i = blockIdx.x * blockDim.x + threadIdx.x;
    if (i < 2 * N_IMG * C_CLS) ws[i] = 0u;
    if (i == 0) ws[WS_MIN] = 0xFFFFFFFFu;
}

__global__ __launch_bounds__(THREADS)
void dl_hist_kernel(const float* __restrict__ logits,
                    const unsigned int* __restrict__ target,
                    unsigned int* __restrict__ ws) {
    __shared__ unsigned int s_pred[C_CLS];
    __shared__ unsigned int s_tgt[C_CLS];
    __shared__ unsigned int s_min;
#if HAVE_ASYNC_LDS
    __shared__ uint4 s_tq[THREADS];
#endif

    const int tid = threadIdx.x;
    if (tid < C_CLS) { s_pred[tid] = 0u; s_tgt[tid] = 0u; }
    if (tid == 0) s_min = 0xFFFFFFFFu;

    const int n     = blockIdx.x / BLOCKS_PER_IMG;
    const int chunk = blockIdx.x % BLOCKS_PER_IMG;
    const size_t q0 = (size_t)chunk * TILE + (size_t)tid * PIX_PER_THREAD; // offset in image
    const float* img = logits + (size_t)n * C_CLS * HW_PIX;
    const unsigned int* gtgt = target + (size_t)n * HW_PIX + q0;

    // --- Stage the 4 int32 targets (16B/lane, coalesced). On gfx1250 use the
    // async global->LDS path (ASYNCcnt) so the fetch overlaps the 19-channel
    // argmax stream below; each lane reads back only its own slot, so only an
    // s_wait_asynccnt is required (no barrier).
#if HAVE_ASYNC_LDS
    __builtin_amdgcn_global_load_async_to_lds_b128(
        (AS_GLOBAL v4i*)(v4i*)gtgt,
        (AS_LDS v4i*)(v4i*)&s_tq[tid],
        /*offset=*/0, /*cpol=*/0);
#else
    const uint4 t4r = *reinterpret_cast<const uint4*>(gtgt);
#endif

    // --- Argmax over 19 channel planes. Logits are a 637MB once-through
    // stream (>> 192MB L2), so use non-temporal loads to keep L2 clean.
    // Strict '>' keeps the first index on ties (matches jnp.argmax).
    // Full unroll: the backend can clause the 18 b128 loads for max MLP.
    f32x4 best = __builtin_nontemporal_load(reinterpret_cast<const f32x4*>(img + q0)); // c = 0
    int bix = 0, biy = 0, biz = 0, biw = 0;
#pragma unroll
    for (int c = 1; c < C_CLS; ++c) {
        const f32x4 v =
            __builtin_nontemporal_load(reinterpret_cast<const f32x4*>(img + (size_t)c * HW_PIX + q0));
        if (v.x > best.x) { best.x = v.x; bix = c; }
        if (v.y > best.y) { best.y = v.y; biy = c; }
        if (v.z > best.z) { best.z = v.z; biz = c; }
        if (v.w > best.w) { best.w = v.w; biw = c; }
    }

#if HAVE_ASYNC_LDS
#if __has_builtin(__builtin_amdgcn_s_wait_asynccnt)
    __builtin_amdgcn_s_wait_asynccnt(0);
#else
    asm volatile("s_wait_asynccnt 0x0" ::: "memory");
#endif
    asm volatile("" ::: "memory"); // keep the LDS read below the wait
    const uint4 t4 = s_tq[tid];
#else
    const uint4 t4 = t4r;
#endif

    __syncthreads(); // s_pred/s_tgt/s_min initialized before any atomics land

    // min over ALL raw target values (reference: target.min(), includes ignored)
    const unsigned int mn = min(min(t4.x, t4.y), min(t4.z, t4.w));
    atomicMin(&s_min, mn);

    // histogram (ignored pixels land in the dropped slot-0 of the reference)
    if (t4.x != IGNORE_IDX) { atomicAdd(&s_pred[bix], 1u); atomicAdd(&s_tgt[t4.x], 1u); }
    if (t4.y != IGNORE_IDX) { atomicAdd(&s_pred[biy], 1u); atomicAdd(&s_tgt[t4.y], 1u); }
    if (t4.z != IGNORE_IDX) { atomicAdd(&s_pred[biz], 1u); atomicAdd(&s_tgt[t4.z], 1u); }
    if (t4.w != IGNORE_IDX) { atomicAdd(&s_pred[biw], 1u); atomicAdd(&s_tgt[t4.w], 1u); }
    __syncthreads();

    if (tid < C_CLS) {
        const unsigned int p = s_pred[tid];
        const unsigned int t = s_tgt[tid];
        if (p) atomicAdd(&ws[WS_PRED + n * C_CLS + tid], p);
        if (t) atomicAdd(&ws[WS_TGT  + n * C_CLS + tid], t);
    }
    if (tid == 0) atomicMin(&ws[WS_MIN], s_min);
}

// Single wave32 finalize: lane = class bin.
__global__ void dl_finalize(const unsigned int* __restrict__ ws, float* __restrict__ out) {
    const int lane = threadIdx.x;            // 0..31, wave32
    const bool active = (lane < C_CLS);
    const unsigned int gmin = ws[WS_MIN];
    const int shift = (gmin == 0u) ? 1 : 0;  // reference: shift = (target.min()==0)

    float loss = 0.0f;
    for (int n = 0; n < N_IMG; ++n) {
        float pc = active ? (float)ws[WS_PRED + n * C_CLS + lane] : 0.0f;
        float tc = active ? (float)ws[WS_TGT  + n * C_CLS + lane] : 0.0f;

        // wave reductions for the per-image sums
        float psum = pc, tsum = tc;
        for (int off = 16; off > 0; off >>= 1) {
            psum += __shfl_down(psum, off, 32);
            tsum += __shfl_down(tsum, off, 32);
        }
        psum = __shfl(psum, 0, 32);
        tsum = __shfl(tsum, 0, 32);

        float t_count, t_den;
        if (shift) {
            // bins 1..C of (target+1) == raw bins 0..C-1
            t_count = tc;
            t_den   = tsum;
        } else {
            // bins 1..C of raw target: bin b takes raw[b+1]; class 0 dropped everywhere
            const float raw0 = __shfl(tc, 0, 32);
            const float nxt  = __shfl_down(tc, 1, 32);   // lane 18 pulls lane 19 (==0)
            t_count = active ? nxt : 0.0f;
            t_den   = tsum - raw0;
        }

        float diff = active ? fabsf(t_count / t_den - pc / psum) : 0.0f;
        for (int off = 16; off > 0; off >>= 1) diff += __shfl_down(diff, off, 32);
        if (lane == 0) loss += diff;
    }
    if (lane == 0) out[0] = loss / (float)N_IMG;
}

extern "C" void kernel_launch(void* const* d_in, const int* in_sizes, int n_in,
                              void* d_out, int out_size, void* d_ws, size_t ws_size,
                              hipStream_t stream) {
    const float*        logits = (const float*)d_in[0];
    const unsigned int* target = (const unsigned int*)d_in[1]; // int32 per harness contract
    unsigned int*       ws     = (unsigned int*)d_ws;
    float*              out    = (float*)d_out;

    dl_init_ws<<<1, 1024, 0, stream>>>(ws);
    dl_hist_kernel<<<N_IMG * BLOCKS_PER_IMG, THREADS, 0, stream>>>(logits, target, ws);
    dl_finalize<<<1, 32, 0, stream>>>(ws, out);
}